// GAT_27144193311437
// MI455X (gfx1250) — compile-verified
//
#include <hip/hip_runtime.h>
#include <hip/hip_bf16.h>

typedef __attribute__((ext_vector_type(2))) float v2f;
typedef __attribute__((ext_vector_type(8))) float v8f;

#define NEG_SLOPE 0.2f

__device__ __forceinline__ unsigned f2ou(float f) {
    unsigned b = __float_as_uint(f);
    return (b & 0x80000000u) ? ~b : (b | 0x80000000u);
}
__device__ __forceinline__ float ou2f(unsigned u) {
    return (u & 0x80000000u) ? __uint_as_float(u & 0x7fffffffu)
                             : __uint_as_float(~u);
}
// ordered-uint encoding of -inf (identity element for the float max)
#define ORDERED_NEG_INF 0x007FFFFFu

// ---------------------------------------------------------------------------
// GEMM: Y[n,o] = sum_k X[n,k] * W[o,k];  X:[nrows,256], W:[256,256] row-major.
//
// Block = 256 threads = 8 wave32s. All waves share one 64-column strip of W
// staged in LDS (two K-chunks of 128; 64 x 132 floats = 33 KB; stride 132
// dwords => each ds_load_b64 across a wave hits all 64 banks exactly once).
// Each wave computes TWO 16-row x 64-col tiles (32 rows): the 4 B fragments
// per K-step are reused by 2 A fragments -> 8 WMMAs per 4 LDS b64 reads,
// i.e. 16384 FLOPs per KB of LDS traffic (2x round-2), keeping the
// V_WMMA_F32_16X16X4_F32 pipe fed instead of the LDS port.
//
// Fragment layouts (ISA 7.12.2, 32-bit):
//   A 16x4 : V0 = K0 (lanes 0-15) / K2 (lanes 16-31); V1 = K1 / K3
//   B 4x16 : V0 = rows K0/K2 split by half-wave;      V1 = K1 / K3
//   C 16x16: VGPR r = row r (lanes 0-15) / row r+8 (lanes 16-31), N = lane&15
// ---------------------------------------------------------------------------
#define LDS_STRIDE 132   // 128 + 4 dword pad: conflict-free & 16B-aligned rows

__global__ void __launch_bounds__(256)
gemm256_wmma_f32(const float* __restrict__ X, const float* __restrict__ W,
                 float* __restrict__ Y, int nrows) {
    __shared__ float ldsW[64 * LDS_STRIDE];   // 33,792 B

    const int tid  = threadIdx.x;
    const int wave = tid >> 5;
    const int lane = tid & 31;
    const int l    = lane & 15;
    const int hi   = lane >> 4;

    const int rowTile = blockIdx.x * 256 + wave * 32;   // per-wave 32 rows
    const int colBase = blockIdx.y * 64;                // per-block 64 cols

    int arow0 = rowTile + l;
    int arow1 = rowTile + 16 + l;
    if (arow0 >= nrows) arow0 = nrows - 1;              // clamp loads, guard stores
    if (arow1 >= nrows) arow1 = nrows - 1;
    const float* xp0 = X + (size_t)arow0 * 256 + 2 * hi;
    const float* xp1 = X + (size_t)arow1 * 256 + 2 * hi;

    v8f c00 = {}, c01 = {}, c02 = {}, c03 = {};         // rows rowTile..+15
    v8f c10 = {}, c11 = {}, c12 = {}, c13 = {};         // rows rowTile+16..+31

    for (int k0 = 0; k0 < 256; k0 += 128) {
        __syncthreads();                                // LDS reuse fence
        // ---- cooperative stage: W[colBase..+63][k0..+127] -> LDS ----
        // 8192 floats = 2048 float4s; 8 per thread, coalesced along k.
#pragma unroll
        for (int j = 0; j < 8; ++j) {
            int i4 = tid + j * 256;                     // float4 index
            int r  = i4 >> 5;                           // 0..63  (col within strip)
            int c4 = i4 & 31;                           // 0..31  (float4 within 128)
            float4 v = *(const float4*)(W + (size_t)(colBase + r) * 256 + k0 + c4 * 4);
            *(float4*)(ldsW + r * LDS_STRIDE + c4 * 4) = v;
        }
        __syncthreads();

        if (k0 == 0) {                                  // warm next A chunk
            __builtin_prefetch(xp0 + 128, 0, 0);
            __builtin_prefetch(xp1 + 128, 0, 0);
        }

#pragma unroll 2
        for (int kk = 0; kk < 128; kk += 4) {
            v2f a0 = { xp0[k0 + kk], xp0[k0 + kk + 1] };
            v2f a1 = { xp1[k0 + kk], xp1[k0 + kk + 1] };
            const float* lp = ldsW + kk + 2 * hi;
            v2f b0 = { lp[(l +  0) * LDS_STRIDE], lp[(l +  0) * LDS_STRIDE + 1] };
            v2f b1 = { lp[(l + 16) * LDS_STRIDE], lp[(l + 16) * LDS_STRIDE + 1] };
            v2f b2 = { lp[(l + 32) * LDS_STRIDE], lp[(l + 32) * LDS_STRIDE + 1] };
            v2f b3 = { lp[(l + 48) * LDS_STRIDE], lp[(l + 48) * LDS_STRIDE + 1] };
            c00 = __builtin_amdgcn_wmma_f32_16x16x4_f32(false, a0, false, b0, (short)0, c00, false, false);
            c01 = __builtin_amdgcn_wmma_f32_16x16x4_f32(false, a0, false, b1, (short)0, c01, false, false);
            c02 = __builtin_amdgcn_wmma_f32_16x16x4_f32(false, a0, false, b2, (short)0, c02, false, false);
            c03 = __builtin_amdgcn_wmma_f32_16x16x4_f32(false, a0, false, b3, (short)0, c03, false, false);
            c10 = __builtin_amdgcn_wmma_f32_16x16x4_f32(false, a1, false, b0, (short)0, c10, false, false);
            c11 = __builtin_amdgcn_wmma_f32_16x16x4_f32(false, a1, false, b1, (short)0, c11, false, false);
            c12 = __builtin_amdgcn_wmma_f32_16x16x4_f32(false, a1, false, b2, (short)0, c12, false, false);
            c13 = __builtin_amdgcn_wmma_f32_16x16x4_f32(false, a1, false, b3, (short)0, c13, false, false);
        }
    }

#pragma unroll
    for (int r = 0; r < 8; ++r) {
        int row0 = rowTile + r + 8 * hi;
        if (row0 < nrows) {
            float* yp = Y + (size_t)row0 * 256 + colBase + l;
            yp[0]  = c00[r];
            yp[16] = c01[r];
            yp[32] = c02[r];
            yp[48] = c03[r];
        }
        int row1 = rowTile + 16 + r + 8 * hi;
        if (row1 < nrows) {
            float* yp = Y + (size_t)row1 * 256 + colBase + l;
            yp[0]  = c10[r];
            yp[16] = c11[r];
            yp[32] = c12[r];
            yp[48] = c13[r];
        }
    }
}

// ---------------------------------------------------------------------------
// Elementwise / graph kernels
// ---------------------------------------------------------------------------
__global__ void fill_f32(float* p, float v, int n) {
    int i = blockIdx.x * blockDim.x + threadIdx.x;
    if (i < n) p[i] = v;
}
__global__ void fill_u32(unsigned* p, unsigned v, int n) {
    int i = blockIdx.x * blockDim.x + threadIdx.x;
    if (i < n) p[i] = v;
}
__global__ void init_bias256(float* p, const float* __restrict__ b, int n) {
    int i = blockIdx.x * blockDim.x + threadIdx.x;
    if (i < n) p[i] = b[i & 255];
}

// self-loop attr: scatter-mean of ew over source nodes
__global__ void loop_stats(const int* __restrict__ src, const float* __restrict__ ew,
                           float* cnt, float* sum, int E) {
    int e = blockIdx.x * blockDim.x + threadIdx.x;
    if (e < E) {
        atomicAdd(&cnt[src[e]], 1.0f);
        atomicAdd(&sum[src[e]], ew[e]);
    }
}
__global__ void loop_div(float* attr, const float* __restrict__ cnt, int n) {
    int i = blockIdx.x * blockDim.x + threadIdx.x;
    if (i < n) attr[i] = attr[i] / fmaxf(cnt[i], 1.0f);
}

// per-node per-head attention logits: a_src[n,h] = <xl[n,h,:], att_s[h,:]>
__global__ void attn_node(const float* __restrict__ xl,
                          const float* __restrict__ att_s,
                          const float* __restrict__ att_d,
                          float* aS, float* aD, int n) {
    int idx = blockIdx.x * blockDim.x + threadIdx.x;  // n*4 + h
    if (idx >= n * 4) return;
    int node = idx >> 2, h = idx & 3;
    const float* v = xl + (size_t)node * 256 + h * 64;
    const float* s = att_s + h * 64;
    const float* d = att_d + h * 64;
    float ss = 0.f, sd = 0.f;
#pragma unroll 8
    for (int c = 0; c < 64; ++c) {
        float x = v[c];
        ss += x * s[c];
        sd += x * d[c];
    }
    aS[idx] = ss;
    aD[idx] = sd;
}

// a_e[e,h] = ew*<We_h,att_e_h> + <be_h,att_e_h>  -> 8 precomputed scalars
__global__ void edge_coef(const float* __restrict__ We, const float* __restrict__ be,
                          const float* __restrict__ att_e, float* kekb) {
    int h = threadIdx.x;
    if (h >= 4) return;
    float ke = 0.f, kb = 0.f;
    for (int c = 0; c < 64; ++c) {
        ke += We[h * 64 + c] * att_e[h * 64 + c];
        kb += be[h * 64 + c] * att_e[h * 64 + c];
    }
    kekb[h]     = ke;
    kekb[4 + h] = kb;
}

// per-edge (incl. virtual self loops at e >= E) leaky-relu logits
__global__ void edge_alpha(const int* __restrict__ src, const int* __restrict__ dst,
                           const float* __restrict__ ew, const float* __restrict__ attr,
                           const float* __restrict__ aS, const float* __restrict__ aD,
                           const float* __restrict__ kekb,
                           float* alpha, int E, int N) {
    int e = blockIdx.x * blockDim.x + threadIdx.x;
    if (e >= E + N) return;
    int s, d; float w;
    if (e < E) { s = src[e]; d = dst[e]; w = ew[e]; }
    else       { s = d = e - E;          w = attr[s]; }
#pragma unroll
    for (int h = 0; h < 4; ++h) {
        float a = aS[s * 4 + h] + aD[d * 4 + h] + w * kekb[h] + kekb[4 + h];
        a = (a > 0.f) ? a : NEG_SLOPE * a;
        alpha[(size_t)e * 4 + h] = a;
    }
}

__global__ void seg_max(const int* __restrict__ dst, const float* __restrict__ alpha,
                        unsigned* m, int E, int N) {
    int i = blockIdx.x * blockDim.x + threadIdx.x;  // (E+N)*4
    if (i >= (E + N) * 4) return;
    int e = i >> 2, h = i & 3;
    int d = (e < E) ? dst[e] : e - E;
    atomicMax(&m[d * 4 + h], f2ou(alpha[i]));
}

__global__ void seg_sum(const int* __restrict__ dst, const float* __restrict__ alpha,
                        const unsigned* __restrict__ m, float* s, int E, int N) {
    int i = blockIdx.x * blockDim.x + threadIdx.x;
    if (i >= (E + N) * 4) return;
    int e = i >> 2, h = i & 3;
    int d = (e < E) ? dst[e] : e - E;
    atomicAdd(&s[d * 4 + h], expf(alpha[i] - ou2f(m[d * 4 + h])));
}

// out[dst, c] += xl[src, c] * att(e, head(c)); one edge per 256-thread block.
// The 4 per-head softmax weights are computed once and shared via LDS.
__global__ void scatter_edges(const int* __restrict__ src, const int* __restrict__ dst,
                              const float* __restrict__ alpha,
                              const unsigned* __restrict__ m, const float* __restrict__ ssum,
                              const float* __restrict__ xl, float* out, int E, int N) {
    __shared__ float att4[4];
    int e = blockIdx.x;
    int c = threadIdx.x;          // 0..255
    int s, d;
    if (e < E) { s = src[e]; d = dst[e]; }
    else       { s = d = e - E; }
    if (c < 4) {
        float a = alpha[(size_t)e * 4 + c];
        att4[c] = expf(a - ou2f(m[d * 4 + c])) / ssum[d * 4 + c];
    }
    __syncthreads();
    float att = att4[c >> 6];
    atomicAdd(&out[(size_t)d * 256 + c], xl[(size_t)s * 256 + c] * att);
}

// final: mean over 4 heads + b2  -> [N,64]
__global__ void finalize_mean(const float* __restrict__ acc, const float* __restrict__ b2,
                              float* out, int n) {
    int i = blockIdx.x * blockDim.x + threadIdx.x;  // n*64
    if (i >= n * 64) return;
    int node = i >> 6, c = i & 63;
    const float* p = acc + (size_t)node * 256 + c;
    float v = p[0] + p[64] + p[128] + p[192];
    out[i] = 0.25f * v + b2[c];
}

// ---------------------------------------------------------------------------
extern "C" void kernel_launch(void* const* d_in, const int* in_sizes, int n_in,
                              void* d_out, int out_size, void* d_ws, size_t ws_size,
                              hipStream_t stream) {
    const float* x     = (const float*)d_in[0];
    const int*   ei    = (const int*)  d_in[1];
    const float* ew    = (const float*)d_in[2];
    const float* W1    = (const float*)d_in[3];
    const float* attS1 = (const float*)d_in[4];
    const float* attD1 = (const float*)d_in[5];
    const float* We1   = (const float*)d_in[6];
    const float* be1   = (const float*)d_in[7];
    const float* attE1 = (const float*)d_in[8];
    const float* b1    = (const float*)d_in[9];
    const float* W2    = (const float*)d_in[10];
    const float* attS2 = (const float*)d_in[11];
    const float* attD2 = (const float*)d_in[12];
    const float* We2   = (const float*)d_in[13];
    const float* be2   = (const float*)d_in[14];
    const float* attE2 = (const float*)d_in[15];
    const float* b2    = (const float*)d_in[16];
    float* out = (float*)d_out;

    const int N  = in_sizes[0] / 256;
    const int E  = in_sizes[2];
    const int EN = E + N;
    const int* src = ei;
    const int* dst = ei + E;

    // ---- workspace carve (256B aligned slices) ----
    char* w = (char*)d_ws;
    auto carve = [&](size_t bytes) -> void* {
        void* p = (void*)w;
        w += (bytes + 255) & ~(size_t)255;
        return p;
    };
    float*    bufA  = (float*)   carve((size_t)N * 256 * 4);  // xl (layer1 then layer2)
    float*    bufB  = (float*)   carve((size_t)N * 256 * 4);  // h / out-accumulator
    float*    aS    = (float*)   carve((size_t)N * 4 * 4);
    float*    aD    = (float*)   carve((size_t)N * 4 * 4);
    unsigned* mMax  = (unsigned*)carve((size_t)N * 4 * 4);
    float*    sSum  = (float*)   carve((size_t)N * 4 * 4);
    float*    alpha = (float*)   carve((size_t)EN * 4 * 4);
    float*    cnt   = (float*)   carve((size_t)N * 4);
    float*    attr  = (float*)   carve((size_t)N * 4);
    float*    kekb  = (float*)   carve(256);

    auto cdiv = [](int a, int b) { return (a + b - 1) / b; };
    const dim3 gemmGrid(cdiv(N, 256), 4);   // 8 waves/block, 32 rows/wave

    // ---- self-loop edge attrs: scatter-mean of ew over src ----
    fill_f32<<<cdiv(N, 256), 256, 0, stream>>>(cnt, 0.f, N);
    fill_f32<<<cdiv(N, 256), 256, 0, stream>>>(attr, 0.f, N);
    loop_stats<<<cdiv(E, 256), 256, 0, stream>>>(src, ew, cnt, attr, E);
    loop_div<<<cdiv(N, 256), 256, 0, stream>>>(attr, cnt, N);

    // ================= layer 1 (concat) =================
    gemm256_wmma_f32<<<gemmGrid, 256, 0, stream>>>(x, W1, bufA, N);
    attn_node<<<cdiv(N * 4, 256), 256, 0, stream>>>(bufA, attS1, attD1, aS, aD, N);
    edge_coef<<<1, 4, 0, stream>>>(We1, be1, attE1, kekb);
    edge_alpha<<<cdiv(EN, 256), 256, 0, stream>>>(src, dst, ew, attr, aS, aD, kekb, alpha, E, N);
    fill_u32<<<cdiv(N * 4, 256), 256, 0, stream>>>(mMax, ORDERED_NEG_INF, N * 4);
    fill_f32<<<cdiv(N * 4, 256), 256, 0, stream>>>(sSum, 0.f, N * 4);
    seg_max<<<cdiv(EN * 4, 256), 256, 0, stream>>>(dst, alpha, mMax, E, N);
    seg_sum<<<cdiv(EN * 4, 256), 256, 0, stream>>>(dst, alpha, mMax, sSum, E, N);
    init_bias256<<<cdiv(N * 256, 256), 256, 0, stream>>>(bufB, b1, N * 256);  // out1 = b1
    scatter_edges<<<EN, 256, 0, stream>>>(src, dst, alpha, mMax, sSum, bufA, bufB, E, N);

    // ================= layer 2 (mean over heads) =================
    gemm256_wmma_f32<<<gemmGrid, 256, 0, stream>>>(bufB, W2, bufA, N);
    attn_node<<<cdiv(N * 4, 256), 256, 0, stream>>>(bufA, attS2, attD2, aS, aD, N);
    edge_coef<<<1, 4, 0, stream>>>(We2, be2, attE2, kekb);
    edge_alpha<<<cdiv(EN, 256), 256, 0, stream>>>(src, dst, ew, attr, aS, aD, kekb, alpha, E, N);
    fill_u32<<<cdiv(N * 4, 256), 256, 0, stream>>>(mMax, ORDERED_NEG_INF, N * 4);
    fill_f32<<<cdiv(N * 4, 256), 256, 0, stream>>>(sSum, 0.f, N * 4);
    seg_max<<<cdiv(EN * 4, 256), 256, 0, stream>>>(dst, alpha, mMax, E, N);
    seg_sum<<<cdiv(EN * 4, 256), 256, 0, stream>>>(dst, alpha, mMax, sSum, E, N);
    fill_f32<<<cdiv(N * 256, 256), 256, 0, stream>>>(bufB, 0.f, N * 256);
    scatter_edges<<<EN, 256, 0, stream>>>(src, dst, alpha, mMax, sSum, bufA, bufB, E, N);

    finalize_mean<<<cdiv(N * 64, 256), 256, 0, stream>>>(bufB, b2, out, N);
}